// PhaseEncoder_81226421502239
// MI455X (gfx1250) — compile-verified
//
#include <hip/hip_runtime.h>

// ---------------------------------------------------------------------------
// PhaseEncoder for MI455X (gfx1250): pure streaming workload, ~384 MB moved.
// Roofline floor ~16.5us @ 23.3 TB/s. No matmul -> no WMMA; optimize the
// memory path instead: NT store hints (write set 344MB > 192MB L2) and the
// CDNA5 async global->LDS staging path (ASYNCcnt) for the phase_bins tiles.
// linspace step is precomputed on the host to avoid a per-thread fp32 divide.
// ---------------------------------------------------------------------------

typedef float v4f __attribute__((ext_vector_type(4)));
// exact parameter type expected by __builtin_amdgcn_global_load_async_to_lds_b128
// (clang prints: '__attribute__((__vector_size__(4 * sizeof(int)))) int __device__ *')
typedef int v4i __attribute__((vector_size(16)));
typedef __attribute__((address_space(1))) v4i* gv4i_p;   // global (AS1)
typedef __attribute__((address_space(3))) v4i* lv4i_p;   // LDS (AS3)

#define TWO_PI_F     6.28318530717958647692f
#define INV_TWO_PI_F 0.15915494309189533577f
#define PI_QUARTER_F 0.78539816339744830962f
#define OMEGA_DT_F   0.25132741228718345f   /* 2*pi*40Hz*0.001s */

__global__ __launch_bounds__(256) void phase_encoder_kernel(
    const float* __restrict__ spikes,       // [B,N]  (row 0 used)
    const float* __restrict__ phase_bins,   // [N,8]
    const float* __restrict__ ref_osc,      // [1]
    const float* __restrict__ last_phases,  // [N]
    float* __restrict__ out_enc,            // [B, N*8]   broadcast of pb
    float* __restrict__ out_phases,         // [N]
    float* __restrict__ out_pb,             // [N,8]
    float* __restrict__ out_ref,            // [1]
    float* __restrict__ out_newlast,        // [N]
    float* __restrict__ out_pw,             // [N,8]
    int n, int batch, float off_step)       // off_step = (pi/4)/(N-1), host-computed
{
    __shared__ float lds_pb[256 * 8];       // 8 KB tile of phase_bins
    const int tid = threadIdx.x;
    const int i   = blockIdx.x * 256 + tid;
    if (i >= n) return;

    // ---- stage phase_bins[i,0:8] (32B/lane) through LDS asynchronously ----
    const float* gsrc = phase_bins + (size_t)i * 8;
    float*       ldst = &lds_pb[tid * 8];
#if __has_builtin(__builtin_amdgcn_global_load_async_to_lds_b128)
    __builtin_amdgcn_global_load_async_to_lds_b128(
        (gv4i_p)gsrc, (lv4i_p)ldst, 0, 0);
    __builtin_amdgcn_global_load_async_to_lds_b128(
        (gv4i_p)(gsrc + 4), (lv4i_p)(ldst + 4), 0, 0);
#else
    *(v4f*)ldst       = *(const v4f*)gsrc;
    *(v4f*)(ldst + 4) = *(const v4f*)(gsrc + 4);
#endif

    // ---- overlap: oscillator advance + per-element phase while DMA flies ----
    float ref = ref_osc[0] + OMEGA_DT_F;
    ref -= floorf(ref * INV_TWO_PI_F) * TWO_PI_F;

    float phase = ref + (float)i * off_step;                // linspace(0, pi/4, N)
    phase -= floorf(phase * INV_TWO_PI_F) * TWO_PI_F;

    const bool  spk  = spikes[i] > 0.0f;                    // first batch row
    int bin = (int)(phase * (INV_TWO_PI_F * 8.0f));
    bin = (bin > 7) ? 7 : bin;
    const float addv = spk ? 1.0f : 0.0f;

    out_phases[i]  = phase;
    out_newlast[i] = spk ? phase : last_phases[i];          // (slice 4B-aligned only)
    if (i == 0) out_ref[0] = ref;

    // phase_weights[i,j] = cos(phase - j*2pi/7); slice starts at elem%4==1
    // -> scalar dword NT stores (still fully coalesced).
    const float pw_step = TWO_PI_F / 7.0f;
    float* pw = out_pw + (size_t)i * 8;
#pragma unroll
    for (int j = 0; j < 8; ++j)
        __builtin_nontemporal_store(__cosf(phase - (float)j * pw_step), pw + j);

    // ---- consume the staged tile ----
#if __has_builtin(__builtin_amdgcn_s_wait_asynccnt)
    __builtin_amdgcn_s_wait_asynccnt(0);
#else
    asm volatile("s_wait_asynccnt 0" ::: "memory");
#endif

    float pbv[8];
#pragma unroll
    for (int j = 0; j < 8; ++j) {
        float v = ldst[j];
        v += (j == bin) ? addv : 0.0f;                      // per-row bin add
        pbv[j] = v * 0.95f;                                 // decay
    }
    const v4f lo = { pbv[0], pbv[1], pbv[2], pbv[3] };
    const v4f hi = { pbv[4], pbv[5], pbv[6], pbv[7] };

    // pb output: 16B-aligned -> B128 NT stores
    float* pbo = out_pb + (size_t)i * 8;
    __builtin_nontemporal_store(lo, (v4f*)pbo);
    __builtin_nontemporal_store(hi, (v4f*)(pbo + 4));

    // broadcast pb to every batch row: 256 MB of write-once data -> NT
    const size_t np = (size_t)n * 8;
    for (int b = 0; b < batch; ++b) {
        float* row = out_enc + (size_t)b * np + (size_t)i * 8;
        __builtin_nontemporal_store(lo, (v4f*)row);
        __builtin_nontemporal_store(hi, (v4f*)(row + 4));
    }
}

extern "C" void kernel_launch(void* const* d_in, const int* in_sizes, int n_in,
                              void* d_out, int out_size, void* d_ws, size_t ws_size,
                              hipStream_t stream) {
    (void)n_in; (void)out_size; (void)d_ws; (void)ws_size;

    const float* spikes = (const float*)d_in[0];   // [B,N]
    const float* pbins  = (const float*)d_in[1];   // [N,P]
    const float* refosc = (const float*)d_in[2];   // [1]
    const float* last   = (const float*)d_in[3];   // [N]

    const int n     = in_sizes[3];                 // N = 1<<20
    const int p     = in_sizes[1] / n;             // P = 8
    const int batch = in_sizes[0] / n;             // B = 8
    const size_t np = (size_t)n * p;

    // linspace(0, pi/4, N) step, computed once on host (kills device divide)
    const float off_step = (n > 1) ? (0.78539816339744830962f / (float)(n - 1)) : 0.0f;

    // outputs concatenated flat, in reference return order
    float* out         = (float*)d_out;
    float* out_enc     = out;                      // [B, N*P]
    float* out_phases  = out_enc + (size_t)batch * np;  // [N]
    float* out_pb      = out_phases + n;           // [N,P]
    float* out_ref     = out_pb + np;              // [1]
    float* out_newlast = out_ref + 1;              // [N]
    float* out_pw      = out_newlast + n;          // [N,P]

    dim3 block(256);
    dim3 grid((n + 255) / 256);
    phase_encoder_kernel<<<grid, block, 0, stream>>>(
        spikes, pbins, refosc, last,
        out_enc, out_phases, out_pb, out_ref, out_newlast, out_pw,
        n, batch, off_step);
}